// PointSampler3D_87110526697714
// MI455X (gfx1250) — compile-verified
//
#include <hip/hip_runtime.h>
#include <math.h>

typedef float v2f __attribute__((ext_vector_type(2)));
typedef float v8f __attribute__((ext_vector_type(8)));

#define N_INV 4096
#define N_VIS 65536
#define K_TOP 8
#define SPLIT 8                     // column chunks per row-tile
#define CHUNK (N_VIS / SPLIT)       // 8192 columns per wave
#define ROW_TILES (N_INV / 16)      // 256
#define FLT_BIG 3.4e38f

// Branchless sorted insert of t (t >= 0) into ascending 8-list s[0..7],
// keeping the 8 smallest. r_i = median3(s_{i-1}, s_i, t); r_0 = min(s_0, t)
// expressed as median3(-1, s_0, t) since all values are >= 0.
// __builtin_amdgcn_fmed3f lowers to a single v_med3 with no canonicalize ops.
#define INSERT8(s, t)                                              \
  do {                                                             \
    _Pragma("unroll")                                              \
    for (int _i = 7; _i >= 1; --_i)                                \
      (s)[_i] = __builtin_amdgcn_fmed3f((s)[_i - 1], (s)[_i], (t)); \
    (s)[0] = __builtin_amdgcn_fmed3f(-1.0f, (s)[0], (t));          \
  } while (0)

// ---------------------------------------------------------------------------
// Kernel 1: one wave (32 threads) per (row-tile, column-chunk).
// WMMA f32 16x16x4 computes the 16x16 dot-product tile; per-lane branchless
// top-8 on squared distance; LDS merge -> per-(row,chunk) sorted top-8.
// ---------------------------------------------------------------------------
__global__ __launch_bounds__(32)
void pointsampler_partial_topk(const float* __restrict__ inv,
                               const float* __restrict__ vis,
                               float* __restrict__ partial) {
  const int wave   = blockIdx.x;
  const int rtile  = wave / SPLIT;        // 0..255
  const int chunk  = wave % SPLIT;        // 0..7
  const int lane   = threadIdx.x;         // 0..31 (wave32)
  const int hf     = lane >> 4;           // lane half: 0 or 1
  const int l16    = lane & 15;

  const int row_base = rtile * 16;

  // ---- A fragment (16x4 f32): lane l16 = row; K = 2*hf + v; K==3 -> 0 pad.
  const float* ap = inv + (size_t)(row_base + l16) * 3;
  const float ax = ap[0], ay = ap[1], az = ap[2];
  v2f afrag;
  afrag.x = hf ? az : ax;
  afrag.y = hf ? 0.0f : ay;

  // ---- |a|^2 for the 8 rows covered by this lane's C fragment (M = 8*hf + v)
  float a2[8];
#pragma unroll
  for (int v = 0; v < 8; ++v) {
    const float* rp = inv + (size_t)(row_base + 8 * hf + v) * 3;
    a2[v] = rp[0] * rp[0] + rp[1] * rp[1] + rp[2] * rp[2];
  }

  // ---- per-lane ascending top-8 lists of squared distance, one per row v
  float s[8][8];
#pragma unroll
  for (int v = 0; v < 8; ++v)
#pragma unroll
    for (int e = 0; e < 8; ++e) s[v][e] = FLT_BIG;

  const int col0 = chunk * CHUNK;

  for (int j = 0; j < CHUNK; j += 16) {
    const int col = col0 + j + l16;       // this lane's column (both halves)
    const float* bp = vis + (size_t)col * 3;
    const float bx = bp[0], by = bp[1], bz = bp[2];
    const float b2 = bx * bx + by * by + bz * bz;

    // B fragment (4x16 f32): lane l16 = col; K = 2*hf + v; K==3 -> 0 pad.
    v2f bfrag;
    bfrag.x = hf ? bz : bx;
    bfrag.y = hf ? 0.0f : by;

    v8f c = {};
    // D[m][n] = sum_k A[m][k]*B[k][n]  (EXEC all-ones here, no divergence)
    c = __builtin_amdgcn_wmma_f32_16x16x4_f32(
        /*neg_a=*/false, afrag, /*neg_b=*/false, bfrag,
        /*c_mod=*/(short)0, c, /*reuse_a=*/false, /*reuse_b=*/false);

#pragma unroll
    for (int v = 0; v < 8; ++v) {
      float d2 = fmaxf(a2[v] + b2 - 2.0f * c[v], 0.0f);
      INSERT8(s[v], d2);
    }
  }

  // ---- within-wave merge: 16 column-slots x 8 entries per row -> top-8
  __shared__ float cand[16][16][K_TOP];   // [row-in-tile][colslot][entry] 8KB
#pragma unroll
  for (int v = 0; v < 8; ++v)
#pragma unroll
    for (int e = 0; e < 8; ++e) cand[8 * hf + v][l16][e] = s[v][e];
  __syncthreads();

  if (hf == 0) {                          // lane r in 0..15 handles row r
    float m[K_TOP];
#pragma unroll
    for (int e = 0; e < K_TOP; ++e) m[e] = FLT_BIG;
    for (int cs = 0; cs < 16; ++cs) {
#pragma unroll
      for (int e = 0; e < K_TOP; ++e) {
        const float t = cand[l16][cs][e];
        INSERT8(m, t);
      }
    }
    float* pp = partial + ((size_t)(row_base + l16) * SPLIT + chunk) * K_TOP;
#pragma unroll
    for (int e = 0; e < K_TOP; ++e) pp[e] = m[e];
  }
}

// ---------------------------------------------------------------------------
// Kernel 2: one wave per row; merge SPLIT*8 = 64 sorted candidates -> top-8,
// sqrt, ascending (matches -top_k(-dist)).
// ---------------------------------------------------------------------------
__global__ __launch_bounds__(32)
void pointsampler_final_topk(const float* __restrict__ partial,
                             float* __restrict__ out) {
  const int row  = blockIdx.x;
  const int lane = threadIdx.x;

  __shared__ float cand[SPLIT * K_TOP];   // 64 floats
  const float* pp = partial + (size_t)row * SPLIT * K_TOP;
  cand[lane]      = pp[lane];
  cand[lane + 32] = pp[lane + 32];
  __syncthreads();

  if (lane == 0) {
    float m[K_TOP];
#pragma unroll
    for (int e = 0; e < K_TOP; ++e) m[e] = FLT_BIG;
    for (int i = 0; i < SPLIT * K_TOP; ++i) {
      const float t = cand[i];
      INSERT8(m, t);
    }
#pragma unroll
    for (int e = 0; e < K_TOP; ++e)
      out[(size_t)row * K_TOP + e] = sqrtf(fmaxf(m[e], 0.0f));
  }
}

extern "C" void kernel_launch(void* const* d_in, const int* in_sizes, int n_in,
                              void* d_out, int out_size, void* d_ws, size_t ws_size,
                              hipStream_t stream) {
  const float* inv = (const float*)d_in[0];   // [4096, 3] f32
  const float* vis = (const float*)d_in[1];   // [65536, 3] f32
  // d_in[2] is k (==8), fixed at compile time as K_TOP.
  float* out = (float*)d_out;                 // [4096, 8] f32
  float* partial = (float*)d_ws;              // [4096][SPLIT][8] f32 = 1 MB

  (void)in_sizes; (void)n_in; (void)out_size; (void)ws_size;

  pointsampler_partial_topk<<<ROW_TILES * SPLIT, 32, 0, stream>>>(inv, vis, partial);
  pointsampler_final_topk<<<N_INV, 32, 0, stream>>>(partial, out);
}